// KNN_18236431139163
// MI455X (gfx1250) — compile-verified
//
#include <hip/hip_runtime.h>
#include <math.h>

#ifndef M_PI
#define M_PI 3.14159265358979323846
#endif

#define SEARCH   5
#define K2       25          // SEARCH*SEARCH
#define CENTER   12          // (K2-1)/2
#define KNN      5
#define NCLASSES 20
#define CUTOFF   1.0f
#define PAD      ((SEARCH - 1) / 2)

struct InvGauss { float w[K2]; };

// ---------------------------------------------------------------------------
// Prep: pack (range, label) into one 8-byte record per pixel; pre-apply the
// "negative range -> +inf" transform. Halves the scattered-gather count in
// the main kernel (25 x B64 instead of 50 x B32), puts range+label of a tap
// in the same cacheline, and warms L2 (maps are 1 MB << 192 MB L2).
// ---------------------------------------------------------------------------
__global__ __launch_bounds__(256) void pack_maps(
    const float* __restrict__ proj_range,
    const int*   __restrict__ proj_argmax,
    float2*      __restrict__ packed,
    int n)
{
    int i = blockIdx.x * blockDim.x + threadIdx.x;
    if (i >= n) return;
    float r = proj_range[i];
    r = (r < 0.0f) ? __builtin_inff() : r;
    float2 v;
    v.x = r;
    v.y = __int_as_float(proj_argmax[i]);
    packed[i] = v;
}

// ---------------------------------------------------------------------------
// Post-gather logic: stable top-KNN selection + cutoff + majority vote.
// Streaming swap-if-strictly-less insertion into 5 sorted slots reproduces
// lax.top_k's lower-index tie-break exactly.
// ---------------------------------------------------------------------------
__device__ __forceinline__ float knn_select_vote(
    const float rv[K2], const int av[K2], float ur, const InvGauss& ig)
{
    float sd[KNN]; int sa[KNN];
#pragma unroll
    for (int k = 0; k < KNN; ++k) { sd[k] = __builtin_inff(); sa[k] = NCLASSES; }

#pragma unroll
    for (int t = 0; t < K2; ++t) {
        float d = (t == CENTER) ? 0.0f : fabsf(rv[t] - ur) * ig.w[t];
        int   a = av[t];
#pragma unroll
        for (int k = 0; k < KNN; ++k) {
            bool  lt = d < sd[k];          // strict: ties keep earlier tap first
            float td = sd[k]; int ta = sa[k];
            sd[k] = lt ? d : sd[k];
            sa[k] = lt ? a : sa[k];
            d = lt ? td : d;
            a = lt ? ta : a;
        }
    }

    // cutoff: distances beyond CUTOFF vote as the "invalid" class (excluded)
#pragma unroll
    for (int k = 0; k < KNN; ++k)
        if (sd[k] > CUTOFF) sa[k] = NCLASSES;

    // majority vote over classes 1..NCLASSES-1, ties -> lowest class,
    // empty vote -> class 1 (argmax of all-zero counts + 1)
    int bestCnt = 0, bestLab = 1;
#pragma unroll
    for (int k = 0; k < KNN; ++k) {
        int L = sa[k];
        int c = 0;
#pragma unroll
        for (int j = 0; j < KNN; ++j) c += (sa[j] == L) ? 1 : 0;
        bool validL = (L >= 1) && (L <= NCLASSES - 1);
        bool better = validL && ((c > bestCnt) || (c == bestCnt && L < bestLab));
        bestCnt = better ? c : bestCnt;
        bestLab = better ? L : bestLab;
    }
    return (float)bestLab;
}

// ---------------------------------------------------------------------------
// Main kernel, packed-map path: 25 clamped B64 gathers per point, fully
// unrolled so the compiler clusters the loads and hides L2 latency across
// 8 wave32s per block. Address math is reduced to base + constant-offset with
// a flat clamp (invalid taps only need a safe address — their value is
// replaced by the zero-padding select). Row/col validity is separable and
// computed once per axis with unsigned range checks.
// ---------------------------------------------------------------------------
__global__ __launch_bounds__(256) void knn_vote_packed(
    const float2* __restrict__ packed,
    const float*  __restrict__ unproj_range,
    const int*    __restrict__ px,
    const int*    __restrict__ py,
    float*        __restrict__ out,
    InvGauss ig, int H, int W, int P)
{
    int p = blockIdx.x * blockDim.x + threadIdx.x;

    // L2 re-warm of the packed map: one global_prefetch_b8 per thread covers
    // the whole 1 MB map across the grid (P == H*W here).
    {
        size_t off      = (size_t)p * sizeof(float2);
        size_t mapBytes = (size_t)H * (size_t)W * sizeof(float2);
        if (off < mapBytes)
            __builtin_prefetch((const char*)packed + off, 0, 3);
    }

    if (p >= P) return;

    int   cx = px[p];
    int   cy = py[p];
    float ur = unproj_range[p];

    // separable validity: one unsigned compare per axis offset
    bool vy[SEARCH], vx[SEARCH];
#pragma unroll
    for (int i = 0; i < SEARCH; ++i) {
        vy[i] = (unsigned)(cy + i - PAD) < (unsigned)H;
        vx[i] = (unsigned)(cx + i - PAD) < (unsigned)W;
    }

    const int base  = cy * W + cx;
    const int HWm1  = H * W - 1;

    float rv[K2]; int av[K2];
#pragma unroll
    for (int t = 0; t < K2; ++t) {
        const int iy = t / SEARCH, ix = t % SEARCH;           // compile-time
        int idx = base + (iy - PAD) * W + (ix - PAD);         // base + const
        idx = idx < 0 ? 0 : (idx > HWm1 ? HWm1 : idx);        // flat clamp
        bool valid = vy[iy] & vx[ix];
        float2 v = packed[idx];                               // B64 gather
        rv[t] = valid ? v.x : 0.0f;                           // unfold zero-pad
        av[t] = valid ? __float_as_int(v.y) : 0;
    }

    out[p] = knn_select_vote(rv, av, ur, ig);
}

// ---------------------------------------------------------------------------
// Fallback path (scratch too small): gather directly from the two source maps.
// ---------------------------------------------------------------------------
__global__ __launch_bounds__(256) void knn_vote_direct(
    const float* __restrict__ proj_range,
    const int*   __restrict__ proj_argmax,
    const float* __restrict__ unproj_range,
    const int*   __restrict__ px,
    const int*   __restrict__ py,
    float*       __restrict__ out,
    InvGauss ig, int H, int W, int P)
{
    int p = blockIdx.x * blockDim.x + threadIdx.x;
    {
        size_t off      = (size_t)p * sizeof(float);
        size_t mapBytes = (size_t)H * (size_t)W * sizeof(float);
        if (off < mapBytes) {
            __builtin_prefetch((const char*)proj_range  + off, 0, 3);
            __builtin_prefetch((const char*)proj_argmax + off, 0, 3);
        }
    }
    if (p >= P) return;

    int   cx = px[p];
    int   cy = py[p];
    float ur = unproj_range[p];

    bool vy[SEARCH], vx[SEARCH];
#pragma unroll
    for (int i = 0; i < SEARCH; ++i) {
        vy[i] = (unsigned)(cy + i - PAD) < (unsigned)H;
        vx[i] = (unsigned)(cx + i - PAD) < (unsigned)W;
    }

    const int base = cy * W + cx;
    const int HWm1 = H * W - 1;

    float rv[K2]; int av[K2];
#pragma unroll
    for (int t = 0; t < K2; ++t) {
        const int iy = t / SEARCH, ix = t % SEARCH;
        int idx = base + (iy - PAD) * W + (ix - PAD);
        idx = idx < 0 ? 0 : (idx > HWm1 ? HWm1 : idx);
        bool valid = vy[iy] & vx[ix];
        float r = proj_range[idx];
        int   a = proj_argmax[idx];
        r = (r < 0.0f) ? __builtin_inff() : r;
        rv[t] = valid ? r : 0.0f;
        av[t] = valid ? a : 0;
    }

    out[p] = knn_select_vote(rv, av, ur, ig);
}

// ---------------------------------------------------------------------------
// Host launcher
// ---------------------------------------------------------------------------
extern "C" void kernel_launch(void* const* d_in, const int* in_sizes, int n_in,
                              void* d_out, int out_size, void* d_ws, size_t ws_size,
                              hipStream_t stream)
{
    const float* proj_range  = (const float*)d_in[0];
    const float* unproj      = (const float*)d_in[1];
    const int*   proj_argmax = (const int*)d_in[2];
    const int*   px          = (const int*)d_in[3];
    const int*   py          = (const int*)d_in[4];
    float*       out         = (float*)d_out;

    const int W  = 2048;                 // fixed by the reference
    const int HW = in_sizes[0];
    const int H  = HW / W;
    const int P  = in_sizes[1];

    // inv-gauss weights: 1 - normalized 2D gaussian (sigma = 1), row-major
    // (dy outer, dx inner), computed in double and rounded once.
    InvGauss ig;
    {
        double g[K2], s = 0.0;
        const double mean = (SEARCH - 1) / 2.0;
        for (int y = 0; y < SEARCH; ++y)
            for (int x = 0; x < SEARCH; ++x) {
                double dx = x - mean, dy = y - mean;
                double v = exp(-(dx * dx + dy * dy) * 0.5) / (2.0 * M_PI);
                g[y * SEARCH + x] = v;
                s += v;
            }
        for (int i = 0; i < K2; ++i) ig.w[i] = (float)(1.0 - g[i] / s);
    }

    dim3 blk(256);
    dim3 grdP((P + 255) / 256);
    size_t needed = (size_t)HW * sizeof(float2);

    if (ws_size >= needed) {
        float2* packed = (float2*)d_ws;
        dim3 grdM((HW + 255) / 256);
        pack_maps<<<grdM, blk, 0, stream>>>(proj_range, proj_argmax, packed, HW);
        knn_vote_packed<<<grdP, blk, 0, stream>>>(packed, unproj, px, py, out,
                                                  ig, H, W, P);
    } else {
        knn_vote_direct<<<grdP, blk, 0, stream>>>(proj_range, proj_argmax, unproj,
                                                  px, py, out, ig, H, W, P);
    }
}